// EGraphSAGE_8701603742216
// MI455X (gfx1250) — compile-verified
//
#include <hip/hip_runtime.h>
#include <hip/hip_bf16.h>
#include <cstdint>

typedef __attribute__((ext_vector_type(2))) float v2f;
typedef __attribute__((ext_vector_type(8))) float v8f;

#define F_EDGE 128
#define HDIM   256

// ---------------- zero workspace (agg + cnt) ----------------
__global__ void k_zero(float* __restrict__ p, long long n) {
    long long i = (long long)blockIdx.x * blockDim.x + threadIdx.x;
    if (i < n) p[i] = 0.0f;
}

// ---------------- scatter-add edge features over src ----------------
__global__ __launch_bounds__(256) void k_scatter(const float* __restrict__ edge_attr,
                                                 const long long* __restrict__ src,
                                                 float* __restrict__ agg,
                                                 float* __restrict__ cnt,
                                                 int E) {
    long long idx = (long long)blockIdx.x * blockDim.x + threadIdx.x;
    int e = (int)(idx >> 7);       // / F_EDGE
    int f = (int)(idx & 127);      // % F_EDGE
    if (e >= E) return;
    int s = (int)src[e];
    atomicAdd(agg + (size_t)s * F_EDGE + f, edge_attr[(size_t)e * F_EDGE + f]);
    if (f == 0) atomicAdd(cnt + s, 1.0f);
}

// ---------------- mean = agg / max(cnt,1), in place ----------------
__global__ __launch_bounds__(256) void k_mean(float* __restrict__ agg,
                                              const float* __restrict__ cnt,
                                              int N) {
    long long idx = (long long)blockIdx.x * blockDim.x + threadIdx.x;
    if (idx >= (long long)N * F_EDGE) return;
    int row = (int)(idx >> 7);
    float c = fmaxf(cnt[row], 1.0f);
    agg[idx] = agg[idx] / c;
}

// ---------------- fused GEMM + bias + sigmoid via V_WMMA_F32_16X16X4_F32 ----
// C[nrows x 256] = sigmoid( [A1 | A2] @ W + b ), A1: nrows x K1, A2: nrows x K2
// One wave computes one 16x16 output tile. wave32; EXEC all-1s (uniform control).
//
// f32 16x4 A layout : lane = M + 16*(K>>1), vgpr = K&1
// f32 4x16 B layout : lane = N + 16*(K>>1), vgpr = K&1  (mirror of A)
// f32 16x16 C layout: vgpr r -> M = r + 8*(lane>=16), N = lane&15
__global__ __launch_bounds__(128) void k_gemm_sig_wmma(
    const float* __restrict__ A1, int K1,
    const float* __restrict__ A2, int K2,
    const float* __restrict__ W,  const float* __restrict__ bias,
    float* __restrict__ out, int nrows)
{
    const int lane = threadIdx.x & 31;
    const int wave = threadIdx.x >> 5;
    const int tilesY = HDIM >> 4;              // 16 col-tiles
    const int tilesX = nrows >> 4;             // 3125 row-tiles (nrows % 16 == 0)
    const int tile = blockIdx.x * 4 + wave;
    if (tile >= tilesX * tilesY) return;       // wave-uniform
    const int tx = tile / tilesY;
    const int ty = tile % tilesY;
    const int m0 = tx << 4;
    const int n0 = ty << 4;

    const int half = lane >> 4;                // 0: K = k,k+1 ; 1: K = k+2,k+3
    const int l15  = lane & 15;
    const int K    = K1 + K2;
    const int row  = m0 + l15;                 // A row owned by this lane
    const int col  = n0 + l15;                 // B/C column owned by this lane

    v8f c = {0.f, 0.f, 0.f, 0.f, 0.f, 0.f, 0.f, 0.f};

    for (int k = 0; k < K; k += 4) {
        const int ka = k + 2 * half;           // K1,K2 multiples of 4 -> no straddle
        const float* arow;
        int kk;
        if (ka < K1) { arow = A1 + (size_t)row * K1; kk = ka; }
        else         { arow = A2 + (size_t)row * K2; kk = ka - K1; }
        v2f a;  a.x = arow[kk];                       a.y = arow[kk + 1];
        v2f bb; bb.x = W[(size_t)ka * HDIM + col];    bb.y = W[(size_t)(ka + 1) * HDIM + col];
        c = __builtin_amdgcn_wmma_f32_16x16x4_f32(
                /*neg_a=*/false, a, /*neg_b=*/false, bb,
                /*c_mod=*/(short)0, c, /*reuse_a=*/false, /*reuse_b=*/false);
    }

    const float bn = bias[col];
#pragma unroll
    for (int r = 0; r < 8; ++r) {
        const int m = m0 + r + 8 * half;
        const float x = c[r] + bn;
        out[(size_t)m * HDIM + col] = 1.0f / (1.0f + __expf(-x));
    }
}

// ---------------- per-edge gather + emb write + logit dot ----------------
__global__ __launch_bounds__(256) void k_edge_out(const float* __restrict__ node,
                                                  const long long* __restrict__ srcI,
                                                  const long long* __restrict__ dstI,
                                                  const float* __restrict__ wf,
                                                  const float* __restrict__ bf,
                                                  float* __restrict__ logits,
                                                  float* __restrict__ embs,
                                                  int E) {
    const int wave = threadIdx.x >> 5;
    const int lane = threadIdx.x & 31;
    const long long e = (long long)blockIdx.x * 8 + wave;
    if (e >= E) return;
    const int s = (int)srcI[e];
    const int d = (int)dstI[e];
    const float* ns = node + (size_t)s * HDIM;
    const float* nd = node + (size_t)d * HDIM;
    float* out = embs + (size_t)e * (2 * HDIM);

    float acc = 0.0f;
    for (int j = lane; j < HDIM; j += 32) {
        const float v1 = ns[j];
        const float v2 = nd[j];
        out[j]        = v1;
        out[HDIM + j] = v2;
        acc += v1 * wf[j] + v2 * wf[HDIM + j];
    }
    for (int off = 16; off > 0; off >>= 1)
        acc += __shfl_down(acc, off, 32);
    if (lane == 0) logits[e] = acc + bf[0];
}

extern "C" void kernel_launch(void* const* d_in, const int* in_sizes, int n_in,
                              void* d_out, int out_size, void* d_ws, size_t ws_size,
                              hipStream_t stream) {
    const float*     edge_attr  = (const float*)d_in[0];
    const long long* edge_index = (const long long*)d_in[1];
    const float*     node_attr  = (const float*)d_in[2];
    const float*     w0 = (const float*)d_in[3];
    const float*     b0 = (const float*)d_in[4];
    const float*     w1 = (const float*)d_in[5];
    const float*     b1 = (const float*)d_in[6];
    const float*     wf = (const float*)d_in[7];
    const float*     bf = (const float*)d_in[8];

    const int E = in_sizes[0] / F_EDGE;   // 800000
    const int N = in_sizes[2] / F_EDGE;   // 50000

    const long long* srcI = edge_index;
    const long long* dstI = edge_index + E;

    // workspace layout (floats): mean[N*128] | cnt[N] | node0[N*256] | node1[N*256]
    float* mean  = (float*)d_ws;
    float* cnt   = mean + (size_t)N * F_EDGE;
    float* node0 = cnt + N;
    float* node1 = node0 + (size_t)N * HDIM;

    float* logits = (float*)d_out;
    float* embs   = logits + E;

    // 1) zero agg + cnt
    {
        long long n = (long long)N * F_EDGE + N;
        int blocks = (int)((n + 255) / 256);
        k_zero<<<blocks, 256, 0, stream>>>(mean, n);
    }
    // 2) scatter-add over src
    {
        long long work = (long long)E * F_EDGE;
        int blocks = (int)((work + 255) / 256);
        k_scatter<<<blocks, 256, 0, stream>>>(edge_attr, srcI, mean, cnt, E);
    }
    // 3) mean
    {
        long long work = (long long)N * F_EDGE;
        int blocks = (int)((work + 255) / 256);
        k_mean<<<blocks, 256, 0, stream>>>(mean, cnt, N);
    }
    // 4) layer 0: sigmoid([node_attr | mean] @ w0 + b0) -> node0
    {
        int tiles = (N >> 4) * (HDIM >> 4);           // 50000 waves
        int blocks = (tiles + 3) / 4;                 // 4 waves / block
        k_gemm_sig_wmma<<<blocks, 128, 0, stream>>>(node_attr, F_EDGE, mean, F_EDGE,
                                                    w0, b0, node0, N);
    }
    // 5) layer 1: sigmoid([node0 | mean] @ w1 + b1) -> node1
    {
        int tiles = (N >> 4) * (HDIM >> 4);
        int blocks = (tiles + 3) / 4;
        k_gemm_sig_wmma<<<blocks, 128, 0, stream>>>(node0, HDIM, mean, F_EDGE,
                                                    w1, b1, node1, N);
    }
    // 6) edge embeddings + logits
    {
        int blocks = (E + 7) / 8;                     // 8 edges (waves) / block
        k_edge_out<<<blocks, 256, 0, stream>>>(node1, srcI, dstI, wf, bf,
                                               logits, embs, E);
    }
}